// GraphConvolutionLayer_5669356831168
// MI455X (gfx1250) — compile-verified
//
#include <hip/hip_runtime.h>

typedef __attribute__((ext_vector_type(2))) float v2f;
typedef __attribute__((ext_vector_type(8))) float v8f;

#define N_NODES 100000
#define N_FEAT  128
#define N_EDGES 1600000

#define LDS_PITCH 132                 // 128 + 4 pad -> lane r hits bank 4r (no conflicts)
#define TILES_PER_BLOCK 25            // 6250 row tiles = 250 blocks * 25
#define GEMM_BLOCKS 250

// ---------------------------------------------------------------------------
// CDNA5 async global->LDS copy (ASYNCcnt-tracked, bypasses VGPRs).
// dst LDS byte offset in a VGPR, 64-bit global address in a VGPR pair.
// ---------------------------------------------------------------------------
__device__ __forceinline__ void async_copy16(unsigned lds_off, const void* gptr) {
    unsigned long long ga = (unsigned long long)(uintptr_t)gptr;
    asm volatile("global_load_async_to_lds_b128 %0, %1, off"
                 :: "v"(lds_off), "v"(ga) : "memory");
}
__device__ __forceinline__ void wait_async0() {
    asm volatile("s_wait_asynccnt 0x0" ::: "memory");
}
__device__ __forceinline__ unsigned lds_offset(const void* p) {
    // LDS aperture: flat addr low 32 bits == LDS byte address
    return (unsigned)(uintptr_t)p;
}

// ---------------------------------------------------------------------------
// Kernel 1: zero the segment-sum accumulator in workspace (float4 stores).
// ---------------------------------------------------------------------------
__global__ void gcn_zero_h(float4* __restrict__ h4) {
    int i = blockIdx.x * blockDim.x + threadIdx.x;
    h4[i] = make_float4(0.f, 0.f, 0.f, 0.f);
}

// ---------------------------------------------------------------------------
// Kernel 2: edge scatter.  One wave32 per edge; lane l covers floats 4l..4l+3.
// features (51 MB) and h (51 MB) are L2-resident on the 192 MB L2, so both
// the random gather and the 204.8M f32 atomics stay on-chip.
// ---------------------------------------------------------------------------
__global__ void gcn_scatter(const float* __restrict__ feat,
                            const int*   __restrict__ src,
                            const int*   __restrict__ dst,
                            float*       __restrict__ h) {
    int e = __builtin_amdgcn_readfirstlane(blockIdx.x * 8 + (threadIdx.x >> 5));
    const int lane = threadIdx.x & 31;

    const int s = src[e];
    const int d = dst[e];

    const float4 v = *(const float4*)(feat + (long)s * N_FEAT + lane * 4);
    float* o = h + (long)d * N_FEAT + lane * 4;
    atomicAdd(o + 0, v.x);
    atomicAdd(o + 1, v.y);
    atomicAdd(o + 2, v.z);
    atomicAdd(o + 3, v.w);
}

// ---------------------------------------------------------------------------
// Kernel 3: out = relu(h @ W^T + b) with V_WMMA_F32_16X16X4_F32.
//
// Persistent blocks: 250 blocks x 25 row tiles.  W (64 KB) staged in LDS once
// per block via async-LDS loads; the 16x128 A tile is double-buffered in LDS
// so tile t+1's async copies overlap tile t's WMMA loop.  LDS rows padded to
// 132 floats for bank-conflict-free ds_load_b64.
//
// Per wave (8 waves = 8 column tiles): D(16x16) accumulated over K=128 in 32
// WMMA steps; A layout lanes 0-15 = {M=lane, K 0/1}, lanes 16-31 = {K 2/3};
// B[k][n] = W[n*128+k] mirrored layout.  C/D: lane<16 -> M=vgpr, N=lane;
// lane>=16 -> M=vgpr+8, N=lane-16.
// ---------------------------------------------------------------------------
__global__ void gcn_gemm_bias_relu(const float* __restrict__ h,
                                   const float* __restrict__ W,
                                   const float* __restrict__ b,
                                   float*       __restrict__ out) {
    extern __shared__ float smem[];
    float* sW = smem;                              // 128 * 132 floats
    float* sA0 = sW + 128 * LDS_PITCH;             // 16 * 132 floats
    float* sA1 = sA0 + 16 * LDS_PITCH;             // 16 * 132 floats

    const int tid    = threadIdx.x;
    const int lane   = tid & 31;
    const int waveId = tid >> 5;                   // 0..7 -> column tile
    const int n0     = waveId * 16;
    const int r      = lane & 15;
    const int kp     = (lane >> 4) << 1;           // 0 or 2
    const int rowT0  = blockIdx.x * TILES_PER_BLOCK;

    // ---- stage all of W into LDS (4096 16B chunks, 16 per thread) ----
    #pragma unroll
    for (int c = 0; c < 16; ++c) {
        int i   = c * 256 + tid;                   // chunk index
        int row = i >> 5;                          // 32 chunks per 128-float row
        int col = (i & 31) << 2;                   // float column
        async_copy16(lds_offset(sW + row * LDS_PITCH + col),
                     W + row * N_FEAT + col);
    }
    // ---- stage A tile 0 (512 chunks, 2 per thread) ----
    {
        const float* aG = h + (long)(rowT0 * 16) * N_FEAT;
        #pragma unroll
        for (int c = 0; c < 2; ++c) {
            int i   = c * 256 + tid;
            int row = i >> 5;
            int col = (i & 31) << 2;
            async_copy16(lds_offset(sA0 + row * LDS_PITCH + col),
                         aG + row * N_FEAT + col);
        }
    }
    wait_async0();
    __syncthreads();

    const float bias = b[n0 + r];

    for (int t = 0; t < TILES_PER_BLOCK; ++t) {
        float* sAcur = (t & 1) ? sA1 : sA0;
        float* sAnxt = (t & 1) ? sA0 : sA1;

        // prefetch next A tile into the other buffer (overlaps WMMA below)
        if (t + 1 < TILES_PER_BLOCK) {
            const float* aG = h + (long)((rowT0 + t + 1) * 16) * N_FEAT;
            #pragma unroll
            for (int c = 0; c < 2; ++c) {
                int i   = c * 256 + tid;
                int row = i >> 5;
                int col = (i & 31) << 2;
                async_copy16(lds_offset(sAnxt + row * LDS_PITCH + col),
                             aG + row * N_FEAT + col);
            }
        }

        // ---- 16x16 tile: 32 x V_WMMA_F32_16X16X4_F32 fed from LDS ----
        const float* aP = sAcur + r * LDS_PITCH + kp;
        const float* bP = sW + (n0 + r) * LDS_PITCH + kp;
        v8f acc = {};
        #pragma unroll
        for (int k0 = 0; k0 < N_FEAT; k0 += 4) {
            v2f a  = *(const v2f*)(aP + k0);
            v2f bb = *(const v2f*)(bP + k0);
            acc = __builtin_amdgcn_wmma_f32_16x16x4_f32(
                false, a, false, bb, (short)0, acc, false, false);
        }

        // ---- epilogue: bias + relu, store ----
        const int m0      = (rowT0 + t) * 16;
        const int rowBase = m0 + ((lane >> 4) << 3);
        #pragma unroll
        for (int v = 0; v < 8; ++v) {
            float x = acc[v] + bias;
            out[(long)(rowBase + v) * N_FEAT + n0 + r] = x > 0.f ? x : 0.f;
        }

        wait_async0();        // my async copies into sAnxt are done
        __syncthreads();      // everyone done copying & done reading sAcur
    }
}

// ---------------------------------------------------------------------------
// Launcher (graph-capture safe: kernels only, all on `stream`).
// d_in order: features, src, dst, W, b.  d_ws holds h (51.2 MB).
// ---------------------------------------------------------------------------
extern "C" void kernel_launch(void* const* d_in, const int* in_sizes, int n_in,
                              void* d_out, int out_size, void* d_ws, size_t ws_size,
                              hipStream_t stream) {
    const float* feat = (const float*)d_in[0];
    const int*   src  = (const int*)  d_in[1];
    const int*   dst  = (const int*)  d_in[2];
    const float* W    = (const float*)d_in[3];
    const float* b    = (const float*)d_in[4];
    float*       out  = (float*)d_out;
    float*       h    = (float*)d_ws;

    // 1) zero accumulator: 3,200,000 float4 stores
    gcn_zero_h<<<(N_NODES * N_FEAT / 4) / 256, 256, 0, stream>>>((float4*)h);

    // 2) scatter-add edges: 8 waves/block, 200000 blocks == 1.6M edges exactly
    gcn_scatter<<<N_EDGES / 8, 256, 0, stream>>>(feat, src, dst, h);

    // 3) persistent WMMA GEMM + bias + relu, W + double-buffered A in LDS
    const size_t lds_bytes = (size_t)(128 * LDS_PITCH + 2 * 16 * LDS_PITCH) * sizeof(float);
    gcn_gemm_bias_relu<<<GEMM_BLOCKS, 256, lds_bytes, stream>>>(h, W, b, out);
}